// ChamferBoundarySDFLoss_47236050322054
// MI455X (gfx1250) — compile-verified
//
#include <hip/hip_runtime.h>
#include <hip/hip_bf16.h>

typedef __attribute__((ext_vector_type(2))) float v2f;
typedef __attribute__((ext_vector_type(8))) float v8f;

#define HH 96
#define WW 96
#define NV (95 * 96)          // vertical candidates  = 9120
#define NHC (96 * 95)         // horizontal candidates = 9120
#define NC (NV + NHC)         // total candidates = 18240 = 16 * 1140
#define NT (NC / 16)          // 1140 tiles, exact
#define EPSF 1e-8f
#define BIGC 1000000.0f
#define DIST2 9.0f            // DIST_THRESHOLD^2

#define WAVES_PER_BLOCK 4     // 1140 / 4 = 285 blocks, exact
#define CHUNK 20              // gt tiles staged per LDS buffer; 1140 / 20 = 57 stages, exact
#define NSTAGES (NT / CHUNK)
#define CHUNK_BYTES (CHUNK * 64 * 4)        // 5120 B per buffer
#define COPIES_PER_THREAD (CHUNK_BYTES / (WAVES_PER_BLOCK * 32 * 8))  // 5 x b64 per thread

// ---- candidate extraction (matches extract_zc exactly, row-major ravel) ----
__device__ __forceinline__ void zc_cand(const float* __restrict__ sdf, int k,
                                        float& r, float& c, bool& m) {
    if (k < NV) {
        int i = k / WW, j = k % WW;
        float v1 = sdf[i * WW + j], v2 = sdf[(i + 1) * WW + j];
        float a = fabsf(v1) / (fabsf(v1) + fabsf(v2) + EPSF);
        r = (v1 == 0.0f) ? (float)i
                         : ((v2 == 0.0f) ? (float)(i + 1) : (float)i + a);
        c = (float)j;
        m = (v1 == 0.0f) || (v2 == 0.0f) || (v1 * v2 < 0.0f);
    } else {
        int k2 = k - NV;
        int i = k2 / (WW - 1), j = k2 % (WW - 1);
        float h1 = sdf[i * WW + j], h2 = sdf[i * WW + j + 1];
        float b = fabsf(h1) / (fabsf(h1) + fabsf(h2) + EPSF);
        c = (h1 == 0.0f) ? (float)j
                         : ((h2 == 0.0f) ? (float)(j + 1) : (float)j + b);
        r = (float)i;
        m = (h1 == 0.0f) || (h2 == 0.0f) || (h1 * h2 < 0.0f);
    }
}

// Pack A tiles (pred) and B tiles (gt) in the wave32 WMMA f32 16x16x4 layout:
//   A (16x4, M=lane%16): lanes 0-15 hold (K0,K1)=(pr,pc); lanes 16-31 hold (K2,K3)=(|p|^2, 1)
//   B (4x16, N=lane%16): lanes 0-15 hold (K0,K1)=(-2gr,-2gc); lanes 16-31 hold (K2,K3)=(1, |g|^2)
// so per-K products sum to  |p|^2 + |g|^2 - 2 p.g  = squared distance (exact, f32).
__global__ void extract_kernel(const float* __restrict__ pred,
                               const float* __restrict__ gt,
                               float* __restrict__ Apack, float* __restrict__ Bpack,
                               float* __restrict__ prp,  float* __restrict__ gp,
                               int* __restrict__ prm) {
    int k = blockIdx.x * blockDim.x + threadIdx.x;
    if (k >= NC) return;
    int tile = k >> 4, rw = k & 15;

    float r, c; bool m;
    zc_cand(pred, k, r, c, m);
    prp[2 * k] = r; prp[2 * k + 1] = c;
    prm[k] = m ? 1 : 0;
    float* at = Apack + tile * 64;
    at[rw * 2 + 0] = r;
    at[rw * 2 + 1] = c;
    at[32 + rw * 2 + 0] = r * r + c * c;
    at[32 + rw * 2 + 1] = 1.0f;

    float gr, gc; bool gm;
    zc_cand(gt, k, gr, gc, gm);
    if (!gm) { gr = BIGC; gc = BIGC; }   // gp = where(gt_m, gt_pts, BIG)
    gp[2 * k] = gr; gp[2 * k + 1] = gc;
    float* bt = Bpack + tile * 64;
    bt[rw * 2 + 0] = -2.0f * gr;
    bt[rw * 2 + 1] = -2.0f * gc;
    bt[32 + rw * 2 + 0] = 1.0f;
    bt[32 + rw * 2 + 1] = gr * gr + gc * gc;
}

// Issue CDNA5 async global->LDS copies (ASYNCcnt-tracked) for one chunk of
// B tiles. All 128 threads cooperate; each issues COPIES_PER_THREAD b64 moves.
__device__ __forceinline__ void stage_async(const float* __restrict__ Bpack,
                                            int chunkIdx, unsigned ldsBase, int tid) {
    unsigned gbase = (unsigned)chunkIdx * (unsigned)CHUNK_BYTES + (unsigned)tid * 8u;
    unsigned lbase = ldsBase + (unsigned)tid * 8u;
#pragma unroll
    for (int p = 0; p < COPIES_PER_THREAD; ++p) {
        unsigned goff = gbase + (unsigned)p * (WAVES_PER_BLOCK * 32 * 8);
        unsigned loff = lbase + (unsigned)p * (WAVES_PER_BLOCK * 32 * 8);
        asm volatile("global_load_async_to_lds_b64 %0, %1, %2"
                     :
                     : "v"(loff), "v"(goff), "s"(Bpack)
                     : "memory");
    }
}

// 4 wave32s per block; wave w owns pred tile blockIdx.x*4+w. B tiles are
// double-buffered in LDS via async copies and broadcast to all 4 waves.
// Per stage: 20 v_wmma_f32_16x16x4_f32 + single-op v_med3 clamp + branchless
// (d2, idx) min updates.
__global__ void __launch_bounds__(WAVES_PER_BLOCK * 32)
argmin_kernel(const float* __restrict__ Apack,
              const float* __restrict__ Bpack,
              float* __restrict__ md2,
              int* __restrict__ idxout) {
    __shared__ float buf[2][CHUNK * 64];

    int tid  = threadIdx.x;
    int lane = tid & 31;
    int wid  = tid >> 5;
    int tile = blockIdx.x * WAVES_PER_BLOCK + wid;

    const float* ap = Apack + tile * 64 + lane * 2;
    v2f A; A.x = ap[0]; A.y = ap[1];

    float best[8]; int bidx[8];
#pragma unroll
    for (int v = 0; v < 8; ++v) { best[v] = 3.4e38f; bidx[v] = 0; }
    int colbase = lane & 15;
    const float inf = __builtin_inff();

    unsigned lds0 = (unsigned)(size_t)&buf[0][0];
    unsigned lds1 = (unsigned)(size_t)&buf[1][0];

    stage_async(Bpack, 0, lds0, tid);

    for (int s = 0; s < NSTAGES; ++s) {
        asm volatile("s_wait_asynccnt 0x0" ::: "memory");  // my async writes to LDS done
        __syncthreads();                                   // everyone's done -> buffers valid
        int cur = s & 1;
        if (s + 1 < NSTAGES)
            stage_async(Bpack, s + 1, (cur ? lds0 : lds1), tid);

        const float* bb = &buf[cur][0];
#pragma unroll
        for (int tt = 0; tt < CHUNK; ++tt) {
            v2f B;
            B.x = bb[tt * 64 + lane * 2 + 0];
            B.y = bb[tt * 64 + lane * 2 + 1];
            v8f cacc = {};
            cacc = __builtin_amdgcn_wmma_f32_16x16x4_f32(
                /*neg_a=*/false, A, /*neg_b=*/false, B,
                /*c_mod=*/(short)0, cacc, /*reuse_a=*/false, /*reuse_b=*/false);
            int col = (s * CHUNK + tt) * 16 + colbase;
#pragma unroll
            for (int v = 0; v < 8; ++v) {
                // max(d2, 0) in one v_med3_num_f32 (d2 is always finite)
                float d2 = __builtin_amdgcn_fmed3f(cacc[v], 0.0f, inf);
                if (d2 < best[v]) { best[v] = d2; bidx[v] = col; }
            }
        }
        __syncthreads();   // all waves done reading buf[cur] before it is restaged
    }

    // lexicographic (d2, idx) min within each 16-lane half; first-index ties
#pragma unroll
    for (int msk = 1; msk < 16; msk <<= 1) {
#pragma unroll
        for (int v = 0; v < 8; ++v) {
            float od = __shfl_xor(best[v], msk, 32);
            int   oi = __shfl_xor(bidx[v], msk, 32);
            if (od < best[v] || (od == best[v] && oi < bidx[v])) {
                best[v] = od; bidx[v] = oi;
            }
        }
    }

    if (lane == 0) {
#pragma unroll
        for (int v = 0; v < 8; ++v) { md2[tile * 16 + v] = best[v]; idxout[tile * 16 + v] = bidx[v]; }
    } else if (lane == 16) {
#pragma unroll
        for (int v = 0; v < 8; ++v) { md2[tile * 16 + 8 + v] = best[v]; idxout[tile * 16 + 8 + v] = bidx[v]; }
    }
}

// central/one-sided gradients, matching compute_normals
__device__ __forceinline__ float gradr(const float* s, int i, int j) {
    if (i == 0)      return s[WW + j] - s[j];
    if (i == HH - 1) return s[(HH - 1) * WW + j] - s[(HH - 2) * WW + j];
    return (s[(i + 1) * WW + j] - s[(i - 1) * WW + j]) * 0.5f;
}
__device__ __forceinline__ float gradc(const float* s, int i, int j) {
    if (j == 0)      return s[i * WW + 1] - s[i * WW];
    if (j == WW - 1) return s[i * WW + WW - 1] - s[i * WW + WW - 2];
    return (s[i * WW + j + 1] - s[i * WW + j - 1]) * 0.5f;
}

// inj = sum_k dot_k * bilinear(pred, p_k)  (scatter-then-dot == gather form)
// pix = sum_k (pr_m ? bilinear(pred, p_k) : 0)
__global__ void finalize_kernel(const float* __restrict__ pred,
                                const float* __restrict__ prp,
                                const int* __restrict__ prm,
                                const float* __restrict__ md2,
                                const int* __restrict__ idx,
                                const float* __restrict__ gp,
                                float* __restrict__ out) {
    int k = blockIdx.x * blockDim.x + threadIdx.x;
    float contrib = 0.0f;
    if (k < NC && prm[k]) {
        float r = prp[2 * k], c = prp[2 * k + 1];
        int r0 = min(max((int)floorf(r), 0), HH - 1);
        int c0 = min(max((int)floorf(c), 0), WW - 1);
        int r1 = min(r0 + 1, HH - 1);
        int c1 = min(c0 + 1, WW - 1);
        float ar = r - (float)r0, ac = c - (float)c0;
        float wa = (1.0f - ar) * (1.0f - ac), wb = (1.0f - ar) * ac;
        float wc = ar * (1.0f - ac),          wd = ar * ac;
        float vals = pred[r0 * WW + c0] * wa + pred[r0 * WW + c1] * wb +
                     pred[r1 * WW + c0] * wc + pred[r1 * WW + c1] * wd;
        contrib = vals;                                  // pixel term
        if (md2[k] <= DIST2) {                           // contrib mask
            float nr  = gradr(pred, r0, c0) * wa + gradr(pred, r0, c1) * wb +
                        gradr(pred, r1, c0) * wc + gradr(pred, r1, c1) * wd;
            float ncl = gradc(pred, r0, c0) * wa + gradc(pred, r0, c1) * wb +
                        gradc(pred, r1, c0) * wc + gradc(pred, r1, c1) * wd;
            float nrm = sqrtf(nr * nr + ncl * ncl) + 1e-8f;
            nr /= nrm; ncl /= nrm;
            int gi = idx[k];
            float dr = gp[2 * gi] - r, dc = gp[2 * gi + 1] - c;
            float dot = (dr * nr + dc * ncl) * 1.0f;     // UPDATE_SCALE
            contrib += dot * vals;                       // inject term
        }
    }
    __shared__ float red[256];
    red[threadIdx.x] = contrib;
    __syncthreads();
    for (int s = blockDim.x >> 1; s > 0; s >>= 1) {
        if (threadIdx.x < (unsigned)s) red[threadIdx.x] += red[threadIdx.x + s];
        __syncthreads();
    }
    if (threadIdx.x == 0) atomicAdd(out, red[0]);
}

__global__ void zero_out_kernel(float* out, int n) {
    int k = blockIdx.x * blockDim.x + threadIdx.x;
    if (k < n) out[k] = 0.0f;
}

extern "C" void kernel_launch(void* const* d_in, const int* in_sizes, int n_in,
                              void* d_out, int out_size, void* d_ws, size_t ws_size,
                              hipStream_t stream) {
    const float* pred = (const float*)d_in[0];  // [1,96,96] f32
    const float* gt   = (const float*)d_in[1];  // [1,96,96] f32
    float* out = (float*)d_out;                 // scalar f32

    auto align256 = [](size_t x) { return (x + 255) & ~(size_t)255; };
    char* ws = (char*)d_ws;
    size_t off = 0;
    float* Apack = (float*)(ws + off); off = align256(off + (size_t)NT * 64 * sizeof(float));
    float* Bpack = (float*)(ws + off); off = align256(off + (size_t)NT * 64 * sizeof(float));
    float* prp   = (float*)(ws + off); off = align256(off + (size_t)NC * 2 * sizeof(float));
    float* gp    = (float*)(ws + off); off = align256(off + (size_t)NC * 2 * sizeof(float));
    int*   prm   = (int*)  (ws + off); off = align256(off + (size_t)NC * sizeof(int));
    float* md2   = (float*)(ws + off); off = align256(off + (size_t)NC * sizeof(float));
    int*   idx   = (int*)  (ws + off); off = align256(off + (size_t)NC * sizeof(int));
    (void)ws_size; (void)in_sizes; (void)n_in;

    zero_out_kernel<<<1, 64, 0, stream>>>(out, out_size);
    extract_kernel<<<(NC + 255) / 256, 256, 0, stream>>>(pred, gt, Apack, Bpack, prp, gp, prm);
    argmin_kernel<<<NT / WAVES_PER_BLOCK, WAVES_PER_BLOCK * 32, 0, stream>>>(Apack, Bpack, md2, idx);
    finalize_kernel<<<(NC + 255) / 256, 256, 0, stream>>>(pred, prp, prm, md2, idx, gp, out);
}